// VisionModel_32152125178171
// MI455X (gfx1250) — compile-verified
//
#include <hip/hip_runtime.h>

#define V_NODES 327696
#define E_EDGES 1966080
#define C_DIM   64
#define G_NUM   4
#define EPS_GN  1e-5f
#define NTILES  (V_NODES / 16)   // 20481 exactly

typedef __attribute__((ext_vector_type(2))) float v2f;
typedef __attribute__((ext_vector_type(8))) float v8f;

// ---- order-preserving float <-> uint mapping for atomic max ----
__device__ __forceinline__ unsigned enc_f(float f) {
    unsigned u = __float_as_uint(f);
    return (u & 0x80000000u) ? ~u : (u | 0x80000000u);
}
__device__ __forceinline__ float dec_f(unsigned k) {
    if (k == 0u) return 0.0f;  // key 0 unreachable by enc_f => "no in-edges" -> 0
    unsigned u = (k & 0x80000000u) ? (k & 0x7FFFFFFFu) : ~k;
    return __uint_as_float(u);
}

// ---- zero a small float region ----
__global__ void zero_kernel(float* __restrict__ p, int n) {
    int i = blockIdx.x * blockDim.x + threadIdx.x;
    if (i < n) p[i] = 0.0f;
}

// ---- per-graph sum / sumsq / count via LDS atomics + one global flush ----
__global__ void stats_kernel(const float* __restrict__ src, const int* __restrict__ batch,
                             float* __restrict__ sums, float* __restrict__ sqs,
                             float* __restrict__ cnts) {
    __shared__ float ls[G_NUM * C_DIM];
    __shared__ float lq[G_NUM * C_DIM];
    __shared__ float lc[G_NUM];
    int tid = threadIdx.x;
    for (int i = tid; i < G_NUM * C_DIM; i += blockDim.x) { ls[i] = 0.0f; lq[i] = 0.0f; }
    if (tid < G_NUM) lc[tid] = 0.0f;
    __syncthreads();

    const long long total = (long long)V_NODES * 16;  // float4 granules
    for (long long idx = (long long)blockIdx.x * blockDim.x + tid; idx < total;
         idx += (long long)gridDim.x * blockDim.x) {
        int v = (int)(idx >> 4);
        int c = ((int)idx & 15) * 4;
        int g = batch[v];
        float4 val = *(const float4*)(src + (size_t)v * C_DIM + c);
        atomicAdd(&ls[g * C_DIM + c + 0], val.x);
        atomicAdd(&ls[g * C_DIM + c + 1], val.y);
        atomicAdd(&ls[g * C_DIM + c + 2], val.z);
        atomicAdd(&ls[g * C_DIM + c + 3], val.w);
        atomicAdd(&lq[g * C_DIM + c + 0], val.x * val.x);
        atomicAdd(&lq[g * C_DIM + c + 1], val.y * val.y);
        atomicAdd(&lq[g * C_DIM + c + 2], val.z * val.z);
        atomicAdd(&lq[g * C_DIM + c + 3], val.w * val.w);
        if (c == 0) atomicAdd(&lc[g], 1.0f);
    }
    __syncthreads();
    for (int i = tid; i < G_NUM * C_DIM; i += blockDim.x) {
        atomicAdd(&sums[i], ls[i]);
        atomicAdd(&sqs[i], lq[i]);
    }
    if (tid < G_NUM) atomicAdd(&cnts[tid], lc[tid]);
}

// ---- fold GraphNorm into per-(g,c) affine:  out = A*x + O ----
__global__ void finalize_kernel(const float* __restrict__ sums, const float* __restrict__ sqs,
                                const float* __restrict__ cnts, const float* __restrict__ w,
                                const float* __restrict__ b, const float* __restrict__ ms,
                                float* __restrict__ Acoef, float* __restrict__ Off) {
    int t = threadIdx.x;
    if (t >= G_NUM * C_DIM) return;
    int g = t >> 6, c = t & 63;
    float cnt  = cnts[g];
    float mean = sums[t] / cnt;
    float m    = ms[c];
    // var = E[(x - ms*mean)^2] = E[x^2] - (2m - m^2) * mean^2
    float var  = sqs[t] / cnt - (2.0f * m - m * m) * mean * mean;
    float inv  = rsqrtf(var + EPS_GN);
    float A    = w[c] * inv;
    Acoef[t]   = A;
    Off[t]     = b[c] - A * m * mean;
}

// ---- apply affine norm; optionally zero-init scatter-max key buffer ----
__global__ void apply_norm_kernel(const float* __restrict__ in, const int* __restrict__ batch,
                                  const float* __restrict__ Acoef, const float* __restrict__ Off,
                                  float* __restrict__ out, unsigned* __restrict__ aggkey) {
    long long idx = (long long)blockIdx.x * blockDim.x + threadIdx.x;
    const long long total = (long long)V_NODES * 16;
    if (idx >= total) return;
    int v = (int)(idx >> 4);
    int c = ((int)idx & 15) * 4;
    int g = batch[v];
    float4 x = *(const float4*)(in + (size_t)v * C_DIM + c);
    float4 A = *(const float4*)(Acoef + g * C_DIM + c);
    float4 O = *(const float4*)(Off + g * C_DIM + c);
    float4 r;
    r.x = fmaf(A.x, x.x, O.x);
    r.y = fmaf(A.y, x.y, O.y);
    r.z = fmaf(A.z, x.z, O.z);
    r.w = fmaf(A.w, x.w, O.w);
    *(float4*)(out + (size_t)v * C_DIM + c) = r;
    if (aggkey) {
        uint4 z = make_uint4(0u, 0u, 0u, 0u);
        *(uint4*)(aggkey + (size_t)v * C_DIM + c) = z;
    }
}

// ---- edge scatter-max: single-instruction u32 atomic max per feature ----
__global__ void scatter_max_kernel(const float* __restrict__ h1, const int* __restrict__ ei,
                                   unsigned* __restrict__ aggkey) {
    long long idx = (long long)blockIdx.x * blockDim.x + threadIdx.x;
    const long long total = (long long)E_EDGES * 16;
    if (idx >= total) return;
    int e = (int)(idx >> 4);
    int c = ((int)idx & 15) * 4;
    int s = ei[e];            // edge_index[0][e]
    int d = ei[E_EDGES + e];  // edge_index[1][e]
    float4 v = *(const float4*)(h1 + (size_t)s * C_DIM + c);
    unsigned* kp = aggkey + (size_t)d * C_DIM + c;
    atomicMax(&kp[0], enc_f(v.x));
    atomicMax(&kp[1], enc_f(v.y));
    atomicMax(&kp[2], enc_f(v.z));
    atomicMax(&kp[3], enc_f(v.w));
}

// ---- fused SAGE GEMM + residual + ReLU via f32 WMMA 16x16x4 ----
// out = relu(x0 + dec(agg) @ Wl + bl + h1 @ Wr), written in place over h1.
__global__ void sage_gemm_kernel(const unsigned* __restrict__ aggkey,
                                 const float* __restrict__ h1, const float* __restrict__ x0,
                                 const float* __restrict__ Wl, const float* __restrict__ bl,
                                 const float* __restrict__ Wr, float* __restrict__ out) {
    __shared__ float sWl[C_DIM * C_DIM];
    __shared__ float sWr[C_DIM * C_DIM];
    __shared__ float sB[C_DIM];
    int tid = threadIdx.x;
    for (int i = tid; i < C_DIM * C_DIM; i += 256) {
        sWl[i] = Wl[i];
        sWr[i] = Wr[i];
    }
    if (tid < C_DIM) sB[tid] = bl[tid];
    __syncthreads();

    int lane  = tid & 31;
    int wave  = tid >> 5;
    int tile  = blockIdx.x * 8 + wave;
    if (tile >= NTILES) return;  // whole wave exits: EXEC stays all-ones for WMMA

    const int m0    = tile * 16;
    const int ml    = lane & 15;
    const int khalf = lane >> 4;             // 0: K+{0,1}, 1: K+{2,3}
    const int mrow  = m0 + ml;               // A-matrix row owned by this lane

    v8f acc[4];
    acc[0] = {}; acc[1] = {}; acc[2] = {}; acc[3] = {};

#pragma unroll 2
    for (int kk = 0; kk < 16; ++kk) {
        int ka = 4 * kk + 2 * khalf;
        // A fragments as explicit 64-bit loads
        // (16x4 f32 layout: lanes 0-15 hold K0/K1, lanes 16-31 K2/K3)
        v2f aR = *(const v2f*)(h1 + (size_t)mrow * C_DIM + ka);
        uint2 kv = *(const uint2*)(aggkey + (size_t)mrow * C_DIM + ka);
        v2f aL; aL.x = dec_f(kv.x); aL.y = dec_f(kv.y);
#pragma unroll
        for (int nt = 0; nt < 4; ++nt) {
            int n = nt * 16 + ml;
            // B fragment (4x16 f32, mirrored K split across lane halves)
            v2f bL; bL.x = sWl[ka * C_DIM + n]; bL.y = sWl[(ka + 1) * C_DIM + n];
            v2f bR; bR.x = sWr[ka * C_DIM + n]; bR.y = sWr[(ka + 1) * C_DIM + n];
            acc[nt] = __builtin_amdgcn_wmma_f32_16x16x4_f32(
                false, aL, false, bL, (short)0, acc[nt], false, false);
            acc[nt] = __builtin_amdgcn_wmma_f32_16x16x4_f32(
                false, aR, false, bR, (short)0, acc[nt], false, false);
        }
    }

    // epilogue: residual + bias + ReLU (C/D layout: VGPR j -> M = j + 8*khalf)
    // Batch the 8 residual loads per N-tile into independent regs so the
    // compiler can issue them all before a single wait (no per-element stalls).
#pragma unroll
    for (int nt = 0; nt < 4; ++nt) {
        int col = nt * 16 + ml;
        const float* xp = x0 + (size_t)(m0 + 8 * khalf) * C_DIM + col;
        float*       op = out + (size_t)(m0 + 8 * khalf) * C_DIM + col;
        float xv[8];
#pragma unroll
        for (int j = 0; j < 8; ++j) xv[j] = xp[(size_t)j * C_DIM];
        float bcol = sB[col];
        float res[8];
#pragma unroll
        for (int j = 0; j < 8; ++j) res[j] = fmaxf(xv[j] + acc[nt][j] + bcol, 0.0f);
#pragma unroll
        for (int j = 0; j < 8; ++j) op[(size_t)j * C_DIM] = res[j];
    }
}

extern "C" void kernel_launch(void* const* d_in, const int* in_sizes, int n_in,
                              void* d_out, int out_size, void* d_ws, size_t ws_size,
                              hipStream_t stream) {
    (void)in_sizes; (void)n_in; (void)out_size; (void)ws_size;
    const float* x    = (const float*)d_in[0];
    const int*   ei   = (const int*)d_in[1];
    const int*   batch= (const int*)d_in[2];
    const float* n1w  = (const float*)d_in[3];
    const float* n1b  = (const float*)d_in[4];
    const float* n1ms = (const float*)d_in[5];
    const float* Wl   = (const float*)d_in[6];
    const float* bl   = (const float*)d_in[7];
    const float* Wr   = (const float*)d_in[8];
    const float* n2w  = (const float*)d_in[9];
    const float* n2b  = (const float*)d_in[10];
    const float* n2ms = (const float*)d_in[11];
    float* out = (float*)d_out;

    char* ws = (char*)d_ws;
    size_t off = 0;
    float*    h      = (float*)(ws + off);    off += (size_t)V_NODES * C_DIM * 4;
    unsigned* aggkey = (unsigned*)(ws + off); off += (size_t)V_NODES * C_DIM * 4;
    float*    stat   = (float*)(ws + off);
    float* sums1 = stat + 0;
    float* sqs1  = stat + 256;
    float* cnt1  = stat + 512;   // 16 floats reserved
    float* sums2 = stat + 528;
    float* sqs2  = stat + 784;
    float* cnt2  = stat + 1040;  // 16 floats reserved
    float* A1    = stat + 1056;
    float* O1    = stat + 1312;
    float* A2    = stat + 1568;
    float* O2    = stat + 1824;

    const int ZN = 1056;  // zero sums/sqs/cnts for both norms
    zero_kernel<<<(ZN + 255) / 256, 256, 0, stream>>>(stat, ZN);

    // GraphNorm 1
    stats_kernel<<<1024, 256, 0, stream>>>(x, batch, sums1, sqs1, cnt1);
    finalize_kernel<<<1, 256, 0, stream>>>(sums1, sqs1, cnt1, n1w, n1b, n1ms, A1, O1);
    const long long ap = (long long)V_NODES * 16;
    apply_norm_kernel<<<(unsigned)((ap + 255) / 256), 256, 0, stream>>>(x, batch, A1, O1, h, aggkey);

    // SAGEConv max-aggregation
    const long long sc = (long long)E_EDGES * 16;
    scatter_max_kernel<<<(unsigned)((sc + 255) / 256), 256, 0, stream>>>(h, ei, aggkey);

    // fused GEMMs + residual + ReLU (in place over h)
    sage_gemm_kernel<<<(NTILES + 7) / 8, 256, 0, stream>>>(aggkey, h, x, Wl, bl, Wr, h);

    // GraphNorm 2 -> d_out
    stats_kernel<<<1024, 256, 0, stream>>>(h, batch, sums2, sqs2, cnt2);
    finalize_kernel<<<1, 256, 0, stream>>>(sums2, sqs2, cnt2, n2w, n2b, n2ms, A2, O2);
    apply_norm_kernel<<<(unsigned)((ap + 255) / 256), 256, 0, stream>>>(h, batch, A2, O2, out, nullptr);
}